// WordEmbedding_13829794693471
// MI455X (gfx1250) — compile-verified
//
#include <hip/hip_runtime.h>

// WordEmbedding row-gather for MI455X (gfx1250).
//
// out[row, :] = weight[tokens[row], :]   row = 0..16383, DIM = 1024 fp32
//
// Pure bandwidth problem (~128 MiB total traffic, ~5.8 us floor @ 23.3 TB/s).
// Strategy:
//   - 1 workgroup (256 threads = 8 wave32 waves) per 4 token rows.
//   - Each thread issues 4 independent global_load_b128 (MLP=4/wave), then
//     4 global_store_b128 th:TH_STORE_NT.
//   - 4 contiguous token indices -> single s_load_b128 on the scalar path.
//   - NT stores: the 64 MB output stream is write-once; keep it out of L2 so
//     the 131 MB weight table (fits in 192 MB L2) stays resident and repeated
//     tokens hit in L2 instead of HBM.
//   - 4096 blocks / 32768 waves total: well under front-end launch-rate
//     limits for a ~6 us kernel, prologue amortized over 4 rows.

#define DIM            1024
#define THREADS        256   // 256 * 16 B = 4096 B = one full row per pass
#define ROWS_PER_BLOCK 4

// Native clang vector type (required by __builtin_nontemporal_store).
typedef float v4f __attribute__((ext_vector_type(4)));

__global__ __launch_bounds__(THREADS) void WordEmbedding_gather_kernel(
    const int*   __restrict__ tokens,   // [n_tokens]
    const float* __restrict__ weight,   // [VOCAB, DIM]
    float*       __restrict__ out,      // [n_tokens, DIM]
    int n_tokens)
{
    const int row0 = blockIdx.x * ROWS_PER_BLOCK;
    const int t    = threadIdx.x;

    if (row0 + ROWS_PER_BLOCK <= n_tokens) {
        // Fast path: 4 full rows.
        v4f v[ROWS_PER_BLOCK];

        // 4 contiguous uniform token loads -> s_load_b128; then 4 independent
        // global_load_b128 in flight before any wait.
        #pragma unroll
        for (int r = 0; r < ROWS_PER_BLOCK; ++r) {
            const int tok = tokens[row0 + r];
            v[r] = reinterpret_cast<const v4f*>(weight + (size_t)tok * DIM)[t];
        }

        // 4 NT stores; same SGPR base, r*4096 B folds into the 24-bit ioffset.
        #pragma unroll
        for (int r = 0; r < ROWS_PER_BLOCK; ++r) {
            __builtin_nontemporal_store(
                v[r], reinterpret_cast<v4f*>(out + (size_t)(row0 + r) * DIM) + t);
        }
    } else {
        // Tail (not hit for n_tokens = 16384, kept for generality).
        for (int r = 0; r < ROWS_PER_BLOCK; ++r) {
            const int row = row0 + r;
            if (row >= n_tokens) break;
            const int tok = tokens[row];
            const v4f x = reinterpret_cast<const v4f*>(weight + (size_t)tok * DIM)[t];
            __builtin_nontemporal_store(
                x, reinterpret_cast<v4f*>(out + (size_t)row * DIM) + t);
        }
    }
}

extern "C" void kernel_launch(void* const* d_in, const int* in_sizes, int n_in,
                              void* d_out, int out_size, void* d_ws, size_t ws_size,
                              hipStream_t stream) {
    const int*   tokens = (const int*)  d_in[0];  // integer input -> const int*
    const float* weight = (const float*)d_in[1];  // [32000, 1024] fp32
    float*       out    = (float*)      d_out;    // [n_tokens, 1024] fp32

    const int n_tokens = in_sizes[0];             // 8 * 2048 = 16384

    dim3 grid((n_tokens + ROWS_PER_BLOCK - 1) / ROWS_PER_BLOCK);
    dim3 block(THREADS);
    WordEmbedding_gather_kernel<<<grid, block, 0, stream>>>(tokens, weight, out, n_tokens);
}